// FindNeighbors_13331578487505
// MI455X (gfx1250) — compile-verified
//
#include <hip/hip_runtime.h>
#include <hip/hip_bf16.h>

typedef __attribute__((ext_vector_type(2))) float v2f;
typedef __attribute__((ext_vector_type(4))) float v4f;
typedef __attribute__((ext_vector_type(8))) float v8f;
typedef __attribute__((ext_vector_type(4))) int   v4i;

#define DIM    256
#define TILE   16
#define PITCH  260          // 256 + 4 floats pad -> conflict-free b64 LDS reads
#define WAVES  2
#define BLK    (WAVES * 32)
#define ROWS_PER_BLOCK (WAVES * TILE)   // 32

#if defined(__AMDGCN__) && __has_builtin(__builtin_amdgcn_global_load_async_to_lds_b128)
#define HAVE_ASYNC_LDS 1
typedef __attribute__((address_space(1))) v4i* gptr_v4i;
typedef __attribute__((address_space(3))) v4i* lptr_v4i;
#else
#define HAVE_ASYNC_LDS 0
#endif

// ---------------------------------------------------------------------------
// Kernel 1: embn[i][:] = emb[i][:] * rsqrt(sum(emb[i]^2) + 256*1e-6)
// (sum(x*x + 1e-6, axis=1) == sum(x^2) + 256e-6 for D=256)
// ---------------------------------------------------------------------------
__global__ void __launch_bounds__(256) normalize_rows(const float* __restrict__ emb,
                                                      float* __restrict__ embn) {
    const int wave = threadIdx.x >> 5;
    const int lane = threadIdx.x & 31;
    const int row  = blockIdx.x * 8 + wave;
    const float* src = emb  + (size_t)row * DIM;
    float*       dst = embn + (size_t)row * DIM;

    float vals[8];
    float s = 0.f;
#pragma unroll
    for (int i = 0; i < 8; ++i) {
        float v = src[lane + 32 * i];
        vals[i] = v;
        s += v * v;
    }
#pragma unroll
    for (int m = 16; m >= 1; m >>= 1) s += __shfl_xor(s, m, 32);
    const float inv = rsqrtf(s + 256.0f * 1e-6f);
#pragma unroll
    for (int i = 0; i < 8; ++i) dst[lane + 32 * i] = vals[i] * inv;
}

// Branchless sorted-triple insert, lowest-index tie-break (matches jnp top_k).
// NOTE: bitwise |/& (not ||/&&) so there is NO short-circuit control flow —
// everything lowers to v_cmp + v_cndmask, no exec-mask branches.
// Invariant v0>=v1>=v2 (ties: i0<i1<i2) implies b0 => b1 => b2, so pure selects.
__device__ __forceinline__ void top3_insert(float c, int ci,
                                            float& v0, int& i0,
                                            float& v1, int& i1,
                                            float& v2, int& i2) {
    const bool b0 = (c > v0) | ((c == v0) & (ci < i0));
    const bool b1 = (c > v1) | ((c == v1) & (ci < i1));
    const bool b2 = (c > v2) | ((c == v2) & (ci < i2));
    const float nv0 = b0 ? c  : v0;
    const int   ni0 = b0 ? ci : i0;
    const float nv1 = b0 ? v0 : (b1 ? c  : v1);
    const int   ni1 = b0 ? i0 : (b1 ? ci : i1);
    const float nv2 = b1 ? v1 : (b2 ? c  : v2);
    const int   ni2 = b1 ? i1 : (b2 ? ci : i2);
    v0 = nv0; i0 = ni0; v1 = nv1; i1 = ni1; v2 = nv2; i2 = ni2;
}

// ---------------------------------------------------------------------------
// Kernel 2: streamed cos-sim row blocks via V_WMMA_F32_16X16X4_F32 (dual
// accumulator chains), async global->LDS B-tile staging, online Z + top-3,
// then weighted gather from the ORIGINAL embeddings.
// ---------------------------------------------------------------------------
__global__ void __launch_bounds__(BLK) neighbors_kernel(const float* __restrict__ embn,
                                                        const float* __restrict__ emb,
                                                        float* __restrict__ out,
                                                        int n) {
    __shared__ float ldsA[ROWS_PER_BLOCK * PITCH];
    __shared__ float ldsB[TILE * PITCH];

    const int tid     = threadIdx.x;
    const int wave    = tid >> 5;
    const int lane    = tid & 31;
    const int g       = lane & 15;   // column within tile / row within A frag
    const int hi      = lane >> 4;   // K-pair select (A/B), row+8 select (C)
    const int rowbase = blockIdx.x * ROWS_PER_BLOCK;

    // ---- stage A tile (32 rows x 256) into LDS, float4 coalesced ----
    const v4f* embn4 = (const v4f*)embn;
    for (int idx = tid; idx < ROWS_PER_BLOCK * (DIM / 4); idx += BLK) {
        int r  = idx >> 6;
        int k4 = idx & 63;
        *(v4f*)&ldsA[r * PITCH + 4 * k4] = embn4[(size_t)(rowbase + r) * (DIM / 4) + k4];
    }

    // ---- per-lane streaming state: 8 rows each ----
    float Z[8], v0[8], v1[8], v2[8];
    int   i0[8], i1[8], i2[8];
#pragma unroll
    for (int r = 0; r < 8; ++r) {
        Z[r] = 0.f;
        v0[r] = v1[r] = v2[r] = -1e30f;
        i0[r] = i1[r] = i2[r] = 0;
    }

    const float* Arow = &ldsA[(wave * TILE + g) * PITCH + 2 * hi];
    const float* Brow = &ldsB[g * PITCH + 2 * hi];

    const int ntiles = n / TILE;
    for (int j = 0; j < ntiles; ++j) {
        __syncthreads();   // previous tile fully consumed before overwrite
        // ---- stage B tile (16 cols x 256): async global->LDS on CDNA5 ----
        for (int idx = tid; idx < TILE * (DIM / 4); idx += BLK) {
            int c  = idx >> 6;
            int k4 = idx & 63;
#if HAVE_ASYNC_LDS
            __builtin_amdgcn_global_load_async_to_lds_b128(
                (gptr_v4i)(embn4 + (size_t)(j * TILE + c) * (DIM / 4) + k4),
                (lptr_v4i)&ldsB[c * PITCH + 4 * k4],
                0, 0);
#else
            *(v4f*)&ldsB[c * PITCH + 4 * k4] =
                embn4[(size_t)(j * TILE + c) * (DIM / 4) + k4];
#endif
        }
#if HAVE_ASYNC_LDS
#if __has_builtin(__builtin_amdgcn_s_wait_asynccnt)
        __builtin_amdgcn_s_wait_asynccnt(0);
#else
        asm volatile("s_wait_asynccnt 0x0" ::: "memory");
#endif
#endif
        __syncthreads();   // B tile visible to both waves

        // ---- 64 x v_wmma_f32_16x16x4_f32, two independent accum chains ----
        v8f acc0 = {0.f, 0.f, 0.f, 0.f, 0.f, 0.f, 0.f, 0.f};
        v8f acc1 = {0.f, 0.f, 0.f, 0.f, 0.f, 0.f, 0.f, 0.f};
#pragma unroll 8
        for (int k = 0; k < DIM; k += 8) {
            v2f a0 = *(const v2f*)(Arow + k);
            v2f b0 = *(const v2f*)(Brow + k);
            v2f a1 = *(const v2f*)(Arow + k + 4);
            v2f b1 = *(const v2f*)(Brow + k + 4);
            acc0 = __builtin_amdgcn_wmma_f32_16x16x4_f32(
                false, a0, false, b0, (short)0, acc0, false, false);
            acc1 = __builtin_amdgcn_wmma_f32_16x16x4_f32(
                false, a1, false, b1, (short)0, acc1, false, false);
        }
        v8f acc = acc0 + acc1;

        const int col = j * TILE + g;
#pragma unroll
        for (int r = 0; r < 8; ++r) {
            float c = acc[r];              // cos in ~[-1,1]: exp safe, no max
            Z[r] += __expf(c);
            top3_insert(c, col, v0[r], i0[r], v1[r], i1[r], v2[r], i2[r]);
        }
    }

    // ---- butterfly merge across the 16 lanes of each half-wave group ----
#pragma unroll
    for (int r = 0; r < 8; ++r) {
#pragma unroll
        for (int m = 1; m <= 8; m <<= 1) {
            Z[r] += __shfl_xor(Z[r], m, 32);
            float ov0 = __shfl_xor(v0[r], m, 32); int oi0 = __shfl_xor(i0[r], m, 32);
            float ov1 = __shfl_xor(v1[r], m, 32); int oi1 = __shfl_xor(i1[r], m, 32);
            float ov2 = __shfl_xor(v2[r], m, 32); int oi2 = __shfl_xor(i2[r], m, 32);
            top3_insert(ov0, oi0, v0[r], i0[r], v1[r], i1[r], v2[r], i2[r]);
            top3_insert(ov1, oi1, v0[r], i0[r], v1[r], i1[r], v2[r], i2[r]);
            top3_insert(ov2, oi2, v0[r], i0[r], v1[r], i1[r], v2[r], i2[r]);
        }
    }

    // ---- weights + gather: row = rowbase + wave*16 + hi*8 + r ----
    // first softmax value at top-k: s_k = exp(c_k)/Z ; second softmax over s_k
#pragma unroll
    for (int r = 0; r < 8; ++r) {
        const float invZ = 1.0f / Z[r];
        const float s0 = __expf(v0[r]) * invZ;
        const float s1 = __expf(v1[r]) * invZ;
        const float s2 = __expf(v2[r]) * invZ;
        const float e0 = __expf(s0), e1 = __expf(s1), e2 = __expf(s2);
        const float inv = 1.0f / (e0 + e1 + e2);
        const float w0 = e0 * inv, w1 = e1 * inv, w2 = e2 * inv;

        const float* p0 = emb + (size_t)i0[r] * DIM;
        const float* p1 = emb + (size_t)i1[r] * DIM;
        const float* p2 = emb + (size_t)i2[r] * DIM;
        const int row = rowbase + wave * TILE + hi * 8 + r;
        float* op = out + (size_t)row * DIM;

#pragma unroll
        for (int q = 0; q < 4; ++q) {
            const int d = g * 16 + q * 4;   // 16 lanes cover 256 cols, coalesced
            v4f a0 = *(const v4f*)(p0 + d);
            v4f a1 = *(const v4f*)(p1 + d);
            v4f a2 = *(const v4f*)(p2 + d);
            v4f o  = a0 * w0 + a1 * w1 + a2 * w2;
            *(v4f*)(op + d) = o;
        }
    }
}

// ---------------------------------------------------------------------------
extern "C" void kernel_launch(void* const* d_in, const int* in_sizes, int n_in,
                              void* d_out, int out_size, void* d_ws, size_t ws_size,
                              hipStream_t stream) {
    (void)n_in; (void)out_size; (void)ws_size;
    const float* emb  = (const float*)d_in[0];
    float*       out  = (float*)d_out;
    float*       embn = (float*)d_ws;           // 16384*256 f32 = 16 MB scratch
    const int n = in_sizes[0] / DIM;            // 16384

    normalize_rows<<<n / 8, 256, 0, stream>>>(emb, embn);
    neighbors_kernel<<<n / ROWS_PER_BLOCK, BLK, 0, stream>>>(embn, emb, out, n);
}